// IdentityUnetBlock_18021682774867
// MI455X (gfx1250) — compile-verified
//
#include <hip/hip_runtime.h>

typedef float v2f __attribute__((ext_vector_type(2)));
typedef float v8f __attribute__((ext_vector_type(8)));

#define LN_EPS 1e-5f
#define NB   16
#define NC   64
#define HW   16384          // 128*128
#define CHW  (NC * HW)      // 1048576 elements per batch
#define NO   64
#define BLOCKS_PER_BATCH 64
#define ELEMS_PER_BLOCK  (CHW / BLOCKS_PER_BATCH)   // 16384 floats

// ---------------- Pass 1a: per-block partial sum / sumsq ----------------
__global__ void ln_reduce_partials(const float* __restrict__ x,
                                   float* __restrict__ ws) {
    const int blk = blockIdx.x;          // 0..1023
    const int b   = blk >> 6;            // batch
    const int sub = blk & 63;            // sub-block within batch
    const float4* p = (const float4*)(x + (size_t)b * CHW +
                                      (size_t)sub * ELEMS_PER_BLOCK);
    float sum = 0.0f, sq = 0.0f;
    // 16384 floats = 4096 float4, 256 threads
    for (int i = threadIdx.x; i < ELEMS_PER_BLOCK / 4; i += 256) {
        float4 v = p[i];
        sum += v.x + v.y + v.z + v.w;
        sq  += v.x * v.x + v.y * v.y + v.z * v.z + v.w * v.w;
    }
    __shared__ float s_sum[256];
    __shared__ float s_sq[256];
    s_sum[threadIdx.x] = sum;
    s_sq[threadIdx.x]  = sq;
    __syncthreads();
    for (int off = 128; off > 0; off >>= 1) {
        if (threadIdx.x < (unsigned)off) {
            s_sum[threadIdx.x] += s_sum[threadIdx.x + off];
            s_sq[threadIdx.x]  += s_sq[threadIdx.x + off];
        }
        __syncthreads();
    }
    if (threadIdx.x == 0) {
        ws[2 * blk + 0] = s_sum[0];
        ws[2 * blk + 1] = s_sq[0];
    }
}

// ---------------- Pass 1b: finalize mean / inv_std per batch ----------------
__global__ void ln_finalize_stats(float* __restrict__ ws) {
    const int b = blockIdx.x;            // 0..15
    const int t = threadIdx.x;           // 0..63
    __shared__ float s_sum[64];
    __shared__ float s_sq[64];
    s_sum[t] = ws[2 * (b * 64 + t) + 0];
    s_sq[t]  = ws[2 * (b * 64 + t) + 1];
    __syncthreads();
    for (int off = 32; off > 0; off >>= 1) {
        if (t < off) {
            s_sum[t] += s_sum[t + off];
            s_sq[t]  += s_sq[t + off];
        }
        __syncthreads();
    }
    if (t == 0) {
        const float n    = (float)CHW;
        const float mean = s_sum[0] / n;
        const float var  = s_sq[0] / n - mean * mean;
        ws[2048 + 2 * b + 0] = mean;
        ws[2048 + 2 * b + 1] = rsqrtf(var + LN_EPS);
    }
}

// ---------------- Pass 2: fused normalize + 1x1-conv GEMM (WMMA) + ReLU ----
// Wave owns one 16-wide spatial tile; loops 4 batches x 16 K-chunks,
// computing all 4 O-tiles (64 outputs) with V_WMMA_F32_16X16X4_F32.
__global__ void __launch_bounds__(256)
fused_ln_conv_relu(const float* __restrict__ x,
                   const float* __restrict__ lnw,
                   const float* __restrict__ lnb,
                   const float* __restrict__ cw,     // [O=64][C=64]
                   const float* __restrict__ cbias,  // [64]
                   const float* __restrict__ stats,  // [16][2] = mean, inv_std
                   float* __restrict__ out) {
    const int lane  = threadIdx.x & 31;
    const int wid   = (blockIdx.x * blockDim.x + threadIdx.x) >> 5; // 0..4095
    const int stile = wid >> 2;          // 0..1023 spatial tile
    const int bg    = wid & 3;           // batch group (4 batches each)
    const int l15   = lane & 15;
    const int hi    = lane >> 4;         // 0/1 half-wave
    const int s     = stile * 16 + l15;  // spatial index 0..16383

    // A operand rows for this lane: rows l15, l15+16, l15+32, l15+48 of cw
    const float* wrow = cw + (size_t)l15 * NC;

    for (int bi = 0; bi < 4; ++bi) {
        const int b      = bg * 4 + bi;
        const float mean = stats[2 * b + 0];
        const float inv  = stats[2 * b + 1];
        const float* xb  = x + (size_t)b * CHW;

        v8f acc0 = {}, acc1 = {}, acc2 = {}, acc3 = {};

#pragma unroll 4
        for (int k = 0; k < 16; ++k) {
            const int cb = 4 * k + 2 * hi;          // channel base for this lane
            const size_t off0 = (size_t)cb * HW + s;
            // B operand: normalized x for channels cb, cb+1 at spatial s
            const float x0 = xb[off0];
            const float x1 = xb[off0 + HW];
            const float w0 = lnw[off0];
            const float w1 = lnw[off0 + HW];
            const float b0 = lnb[off0];
            const float b1 = lnb[off0 + HW];
            v2f bv;
            bv.x = (x0 - mean) * inv * w0 + b0;
            bv.y = (x1 - mean) * inv * w1 + b1;
            // A operand: conv_w[o][cb..cb+1] for 4 O-tiles (8B aligned, L0-hot)
            const v2f a0 = *(const v2f*)(wrow + cb);
            const v2f a1 = *(const v2f*)(wrow + 16 * NC + cb);
            const v2f a2 = *(const v2f*)(wrow + 32 * NC + cb);
            const v2f a3 = *(const v2f*)(wrow + 48 * NC + cb);

            acc0 = __builtin_amdgcn_wmma_f32_16x16x4_f32(
                false, a0, false, bv, (short)0, acc0, false, false);
            acc1 = __builtin_amdgcn_wmma_f32_16x16x4_f32(
                false, a1, false, bv, (short)0, acc1, false, false);
            acc2 = __builtin_amdgcn_wmma_f32_16x16x4_f32(
                false, a2, false, bv, (short)0, acc2, false, false);
            acc3 = __builtin_amdgcn_wmma_f32_16x16x4_f32(
                false, a3, false, bv, (short)0, acc3, false, false);
        }

        // Store: D VGPR v, lanes 0-15 -> row o=v (64B run), lanes 16-31 -> o=v+8
        float* ob = out + (size_t)b * (size_t)(NO * HW) + s;
#pragma unroll
        for (int v = 0; v < 8; ++v) {
            const int o0 = v + 8 * hi;
            float r;
            r = acc0[v] + cbias[o0];
            ob[(size_t)o0 * HW] = fmaxf(r, 0.0f);
            r = acc1[v] + cbias[16 + o0];
            ob[(size_t)(16 + o0) * HW] = fmaxf(r, 0.0f);
            r = acc2[v] + cbias[32 + o0];
            ob[(size_t)(32 + o0) * HW] = fmaxf(r, 0.0f);
            r = acc3[v] + cbias[48 + o0];
            ob[(size_t)(48 + o0) * HW] = fmaxf(r, 0.0f);
        }
    }
}

extern "C" void kernel_launch(void* const* d_in, const int* in_sizes, int n_in,
                              void* d_out, int out_size, void* d_ws, size_t ws_size,
                              hipStream_t stream) {
    const float* x   = (const float*)d_in[0];
    const float* lnw = (const float*)d_in[1];
    const float* lnb = (const float*)d_in[2];
    const float* cw  = (const float*)d_in[3];
    const float* cb  = (const float*)d_in[4];
    float* out = (float*)d_out;
    float* ws  = (float*)d_ws;   // [0..2047] partials, [2048..2079] stats

    ln_reduce_partials<<<NB * BLOCKS_PER_BATCH, 256, 0, stream>>>(x, ws);
    ln_finalize_stats<<<NB, 64, 0, stream>>>(ws);
    // 1024 spatial tiles x 4 batch-groups = 4096 waves; 8 waves/block
    fused_ln_conv_relu<<<512, 256, 0, stream>>>(x, lnw, lnb, cw, cb,
                                                ws + 2048, out);
}